// BaseModel_81509889344081
// MI455X (gfx1250) — compile-verified
//
#include <hip/hip_runtime.h>

// Pairwise margin-ranking loss via V_WMMA_F32_16X16X4_F32 (rank-4 bilinear form)
// + weighted BCE-with-logits, on gfx1250 (wave32).
//
// prod[i,j] = (p_i - p_j)(l_i - l_j) = [p_i, l_i, p_i*l_i, 1] . [-l_j, -p_j, 1, p_j*l_j]
// -> one 16x16 tile of pairwise products per v_wmma_f32_16x16x4_f32 (exact f32).
// Off-diagonal tiles (tj > ti) have i < j everywhere, so the epilogue is just
// acc += fmaxf(m - prod, 0). Only the diagonal tile needs the i<j mask.

typedef __attribute__((ext_vector_type(2))) float v2f;
typedef __attribute__((ext_vector_type(8))) float v8f;

#define TILE            16
#define MARGIN_THREADS  256
#define WAVES_PER_BLOCK (MARGIN_THREADS / 32)
#define BCE_THREADS     512
#define BCE_BLOCKS      16
#define FIN_THREADS     512

__device__ __forceinline__ float wave_reduce_sum(float v) {
#pragma unroll
    for (int off = 16; off > 0; off >>= 1)
        v += __shfl_xor(v, off, 32);
    return v;
}

__device__ __forceinline__ v8f wmma_prod_tile(v2f a, v2f b) {
    v8f c = {0.f, 0.f, 0.f, 0.f, 0.f, 0.f, 0.f, 0.f};
    // args: (neg_a, A, neg_b, B, c_mod, C, reuse_a, reuse_b)
    return __builtin_amdgcn_wmma_f32_16x16x4_f32(
        false, a, false, b, (short)0, c, false, false);
}

__global__ __launch_bounds__(MARGIN_THREADS)
void margin_tiles_kernel(const float* __restrict__ preds,
                         const float* __restrict__ labels,
                         const int*   __restrict__ margin_p,
                         float* __restrict__ partials,
                         int ntiles)
{
    const int ti     = blockIdx.x;
    const int lane   = threadIdx.x & 31;
    // Force wave id into an SGPR so the tj loop is scalar (s_cmp/s_cbranch),
    // keeping EXEC all-1s statically for the WMMAs.
    const int wave   = __builtin_amdgcn_readfirstlane(threadIdx.x >> 5);
    const int lane16 = lane & 15;
    const bool hi    = lane >= 16;           // lanes 16-31 carry K=2,3 / M=8..15
    const float m    = (float)margin_p[0];

    // A-matrix (16x4, f32): row i = [p_i, l_i, p_i*l_i, 1]
    // Layout: lanes 0-15 -> (K0,K1), lanes 16-31 -> (K2,K3), M = lane&15.
    const int   i_row = ti * TILE + lane16;
    const float pi = preds[i_row];
    const float li = labels[i_row];
    v2f a;
    a.x = hi ? pi * li : pi;    // K=2 : K=0
    a.y = hi ? 1.0f    : li;    // K=3 : K=1

    float acc = 0.0f;

    // ---- diagonal tile (tj == ti): needs the i<j mask; wave 0 only ----
    if (wave == 0) {
        v2f b;
        b.x = hi ? 1.0f    : -li;   // column j uses the same row data
        b.y = hi ? pi * li : -pi;
        v8f d = wmma_prod_tile(a, b);
        const int m_base = hi ? 8 : 0;
#pragma unroll
        for (int r = 0; r < 8; ++r) {
            const float relu = fmaxf(m - d[r], 0.0f);
            if (m_base + r < lane16) acc += relu;   // strict upper triangle
        }
    }

    // ---- off-diagonal tiles (tj > ti): i<j always holds; pure relu ----
    for (int tj = ti + 1 + wave; tj < ntiles; tj += WAVES_PER_BLOCK) {
        // B-matrix (4x16, f32): column j = [-l_j, -p_j, 1, p_j*l_j]
        // N = lane&15; lanes 0-15 -> (K0,K1), lanes 16-31 -> (K2,K3).
        const int   j_col = tj * TILE + lane16;
        const float pj = preds[j_col];
        const float lj = labels[j_col];
        v2f b;
        b.x = hi ? 1.0f    : -lj;   // K=2 : K=0
        b.y = hi ? pj * lj : -pj;   // K=3 : K=1

        v8f d = wmma_prod_tile(a, b);
#pragma unroll
        for (int r = 0; r < 8; ++r)
            acc += fmaxf(m - d[r], 0.0f);
    }

    acc = wave_reduce_sum(acc);
    __shared__ float wsum[WAVES_PER_BLOCK];
    if (lane == 0) wsum[wave] = acc;
    __syncthreads();
    if (threadIdx.x == 0) {
        float s = 0.0f;
#pragma unroll
        for (int w = 0; w < WAVES_PER_BLOCK; ++w) s += wsum[w];
        partials[blockIdx.x] = s;
    }
}

__global__ __launch_bounds__(BCE_THREADS)
void bce_kernel(const float* __restrict__ logits,
                const float* __restrict__ targets,
                const float* __restrict__ pos_weight,
                float* __restrict__ partials,
                int n)
{
    const int idx = blockIdx.x * BCE_THREADS + threadIdx.x;
    const float pw = pos_weight[0];
    float v = 0.0f;
    if (idx < n) {
        const float x = logits[idx];
        const float t = targets[idx];
        const float max_val = fmaxf(-x, 0.0f);
        const float lw = 1.0f + (pw - 1.0f) * t;
        v = (1.0f - t) * x
          + lw * (logf(expf(-max_val) + expf(-x - max_val)) + max_val);
    }

    v = wave_reduce_sum(v);
    __shared__ float wsum[BCE_THREADS / 32];
    const int lane = threadIdx.x & 31;
    const int wave = threadIdx.x >> 5;
    if (lane == 0) wsum[wave] = v;
    __syncthreads();
    if (threadIdx.x == 0) {
        float s = 0.0f;
#pragma unroll
        for (int w = 0; w < BCE_THREADS / 32; ++w) s += wsum[w];
        partials[blockIdx.x] = s;
    }
}

// Single-block deterministic tree reduction of both partial arrays.
__global__ __launch_bounds__(FIN_THREADS)
void finalize_kernel(const float* __restrict__ margin_partials,
                     const float* __restrict__ bce_partials,
                     float* __restrict__ out,
                     int n_margin, int n_bce, float inv_b)
{
    __shared__ float sm[FIN_THREADS];
    const int t = threadIdx.x;

    sm[t] = (t < n_margin) ? margin_partials[t] : 0.0f;
    __syncthreads();
    for (int s = FIN_THREADS / 2; s > 0; s >>= 1) {
        if (t < s) sm[t] += sm[t + s];
        __syncthreads();
    }
    if (t == 0) out[0] = sm[0] * inv_b;
    __syncthreads();

    sm[t] = (t < n_bce) ? bce_partials[t] : 0.0f;
    __syncthreads();
    for (int s = FIN_THREADS / 2; s > 0; s >>= 1) {
        if (t < s) sm[t] += sm[t + s];
        __syncthreads();
    }
    if (t == 0) out[1] = sm[0] * inv_b;
}

extern "C" void kernel_launch(void* const* d_in, const int* in_sizes, int n_in,
                              void* d_out, int out_size, void* d_ws, size_t ws_size,
                              hipStream_t stream) {
    const float* preds      = (const float*)d_in[0];
    const float* labels     = (const float*)d_in[1];
    const float* logits     = (const float*)d_in[2];
    const float* targets    = (const float*)d_in[3];
    const float* pos_weight = (const float*)d_in[4];
    const int*   margin     = (const int*)d_in[5];

    const int n      = in_sizes[0];        // 8192
    const int ntiles = n / TILE;           // 512

    float* ws              = (float*)d_ws;
    float* margin_partials = ws;           // [ntiles]
    float* bce_partials    = ws + ntiles;  // [BCE_BLOCKS]

    margin_tiles_kernel<<<ntiles, MARGIN_THREADS, 0, stream>>>(
        preds, labels, margin, margin_partials, ntiles);

    bce_kernel<<<BCE_BLOCKS, BCE_THREADS, 0, stream>>>(
        logits, targets, pos_weight, bce_partials, n);

    finalize_kernel<<<1, FIN_THREADS, 0, stream>>>(
        margin_partials, bce_partials, (float*)d_out,
        ntiles, BCE_BLOCKS, 1.0f / (float)n);
}